// LIFNeuron_850403525170
// MI455X (gfx1250) — compile-verified
//
#include <hip/hip_runtime.h>
#include <stdint.h>

// ---------------- LIF parameters (match reference) ----------------
#define T_STEPS   128
#define ALPHA_F   0.95122942450071400910f   // exp(-0.001/0.020)
#define BETA_F    0.81873075307798185867f   // exp(-0.001/0.005)

// ---------------- tiling ----------------
#define THREADS       128                    // 4 waves (wave32)
#define ROWS          128                    // neurons per block (1 per thread)
#define ROWS_PER_WAVE 32                     // each wave TDMs its own quarter
#define ROW_DW        (T_STEPS + 2)          // 130 dwords: stride==2 mod 4 -> conflict-free b64 LDS reads
#define ROW_BYTES     (ROW_DW * 4)           // 520 B (8B aligned for b64)
#define HALF_BYTES    (64 * 4)               // 256 B: first 64 timesteps of a row
#define MASK_STRIDE   5                      // odd dword stride -> conflict-free
#define LDS_IN_BYTES   (ROWS * ROW_BYTES)            // 66560
#define LDS_MASK_BYTES (ROWS * MASK_STRIDE * 4)      // 2560
#define LDS_TOTAL      (LDS_IN_BYTES + LDS_MASK_BYTES)

typedef uint32_t u32x4 __attribute__((ext_vector_type(4)));
typedef uint32_t u32x8 __attribute__((ext_vector_type(8)));
typedef float    f32x4 __attribute__((ext_vector_type(4)));

struct __align__(8) f2 { float x, y; };

__global__ __launch_bounds__(THREADS)
void lif_scan_kernel(const float* __restrict__ in, float* __restrict__ out, int n_neurons)
{
    extern __shared__ char smem[];                    // [0, LDS_IN_BYTES): input tile (padded rows)
                                                      // [LDS_IN_BYTES, +): spike bitmasks
    const int tid = threadIdx.x;
    const size_t n0 = (size_t)blockIdx.x * ROWS;      // first neuron of this tile
    if ((int)n0 >= n_neurons) return;

    // LDS byte offset of dynamic-shared base (low 32 bits of generic LDS ptr).
    const uint32_t lds_in_base = (uint32_t)(uintptr_t)(void*)smem;

    // ---------------------------------------------------------------
    // Phase 1: Tensor Data Mover, двойной buffered along T.
    // Each wave issues TWO descriptors for its own 32-row quarter:
    //   desc A: t 0..63   of all 32 rows
    //   desc B: t 64..127 of all 32 rows (LDS dest +256 B)
    // pad_interval=5 (64 dw) + pad_amount=65 (66 dw) interleaves the two
    // half-tiles into the same 130-dword padded rows (conflict-free b64
    // reads).  TDM ops of a wave complete in order, so tensorcnt<=1
    // releases the first half while the second is still in flight.
    // Thread tid's row belongs to its own wave's descriptors, so no
    // cross-wave barrier is needed before compute.
    // ---------------------------------------------------------------
    {
        const uint32_t wid  = (uint32_t)__builtin_amdgcn_readfirstlane(tid >> 5); // wave id, scalar
        const uint64_t ga   = (uint64_t)(uintptr_t)(in + (n0 + wid * ROWS_PER_WAVE) * T_STEPS);
        const uint64_t gb   = ga + HALF_BYTES;
        const uint32_t la   = lds_in_base + wid * (ROWS_PER_WAVE * ROW_BYTES);
        const uint32_t lb   = la + HALF_BYTES;

        // Shared D# group 1:
        //  word0: data_size=2(4B)<<16 | pad_enable<<20 | pad_interval=5(64dw)<<22
        //         | pad_amount=65(66dw)<<25  = 0x83520000
        //  tensor_dim0=64, tensor_dim1=32, tile_dim0=64, tile_dim1=32,
        //  tensor_dim0_stride=128 elements (full row; stride>dim is legal)
        u32x8 g1;
        g1.s0 = 0x83520000u;
        g1.s1 = 64u << 16;          // tensor_dim0 lo16 in [63:48]
        g1.s2 = 32u << 16;          // tensor_dim1 lo16 in [95:80]
        g1.s3 = 64u << 16;          // tile_dim0 in [127:112]
        g1.s4 = 32u;                // tile_dim1 in [143:128]
        g1.s5 = (uint32_t)T_STEPS;  // tensor_dim0_stride lo32
        g1.s6 = 0u;
        g1.s7 = 0u;

        u32x4 gA, gB;
        gA.x = 1u; gA.y = la; gA.z = (uint32_t)ga;
        gA.w = ((uint32_t)(ga >> 32) & 0x01FFFFFFu) | 0x80000000u;   // ga[56:32] | type=2<<30
        gB.x = 1u; gB.y = lb; gB.z = (uint32_t)gb;
        gB.w = ((uint32_t)(gb >> 32) & 0x01FFFFFFu) | 0x80000000u;

        asm volatile("tensor_load_to_lds %0, %1" :: "s"(gA), "s"(g1) : "memory");
        asm volatile("tensor_load_to_lds %0, %1" :: "s"(gB), "s"(g1) : "memory");
    }

    // ---------------------------------------------------------------
    // Phase 2: per-thread LIF recurrence over T, reading its padded LDS
    // row with conflict-free ds_load_b64.  Spikes packed into 4 bitmask
    // dwords (compile-time shifts).  First half runs as soon as desc A
    // lands (tensorcnt<=1); second half after tensorcnt==0.
    // ---------------------------------------------------------------
    {
        const f2* row = (const f2*)(smem + (size_t)tid * ROW_BYTES);
        uint32_t* mask = (uint32_t*)(smem + LDS_IN_BYTES) + tid * MASK_STRIDE;

        float m = 0.0f, s = 0.0f;

        auto run_half = [&](int wbase) {
            #pragma unroll 1
            for (int w = wbase; w < wbase + 2; ++w) {     // 32 timesteps per mask word
                uint32_t bits = 0u;
                #pragma unroll
                for (int j = 0; j < 16; ++j) {            // 2 timesteps per b64 load
                    f2 xv = row[w * 16 + j];
                    s = BETA_F * s + xv.x;                // t = 32w + 2j
                    m = ALPHA_F * m + s;
                    bool sp0 = (m >= 1.0f);
                    m -= sp0 ? 1.0f : 0.0f;
                    bits |= (sp0 ? 1u : 0u) << (2 * j);
                    s = BETA_F * s + xv.y;                // t = 32w + 2j + 1
                    m = ALPHA_F * m + s;
                    bool sp1 = (m >= 1.0f);
                    m -= sp1 ? 1.0f : 0.0f;
                    bits |= (sp1 ? 1u : 0u) << (2 * j + 1);
                }
                mask[w] = bits;
            }
        };

        asm volatile("s_wait_tensorcnt 0x1" ::: "memory");   // desc A landed
        run_half(0);                                          // t 0..63
        asm volatile("s_wait_tensorcnt 0x0" ::: "memory");   // desc B landed
        run_half(2);                                          // t 64..127
    }
    __syncthreads();   // masks of all waves visible for the expansion phase

    // ---------------------------------------------------------------
    // Phase 3: expand bitmasks -> float 0/1 and store with perfectly
    // coalesced non-temporal 16B stores (write-once stream).
    // ---------------------------------------------------------------
    {
        const uint32_t* masks = (const uint32_t*)(smem + LDS_IN_BYTES);
        f32x4* out4 = (f32x4*)(out + n0 * T_STEPS);
        #pragma unroll 4
        for (int i = 0; i < (ROWS * T_STEPS) / (4 * THREADS); ++i) {   // 32 iterations
            int q    = i * THREADS + tid;       // float4 index: row r = q/32, t0 = (q%32)*4
            int r    = q >> 5;
            int lane = q & 31;
            uint32_t mw = masks[r * MASK_STRIDE + (lane >> 3)];
            int bit = (lane & 7) * 4;
            f32x4 v;
            v.x = ((mw >> (bit + 0)) & 1u) ? 1.0f : 0.0f;
            v.y = ((mw >> (bit + 1)) & 1u) ? 1.0f : 0.0f;
            v.z = ((mw >> (bit + 2)) & 1u) ? 1.0f : 0.0f;
            v.w = ((mw >> (bit + 3)) & 1u) ? 1.0f : 0.0f;
            __builtin_nontemporal_store(v, &out4[q]);
        }
    }
}

extern "C" void kernel_launch(void* const* d_in, const int* in_sizes, int n_in,
                              void* d_out, int out_size, void* d_ws, size_t ws_size,
                              hipStream_t stream)
{
    (void)n_in; (void)out_size; (void)d_ws; (void)ws_size;
    const float* in  = (const float*)d_in[0];
    float*       out = (float*)d_out;

    const int total     = in_sizes[0];          // B*F*T
    const int n_neurons = total / T_STEPS;      // B*F = 524288
    const int blocks    = n_neurons / ROWS;     // 4096

    lif_scan_kernel<<<blocks, THREADS, LDS_TOTAL, stream>>>(in, out, n_neurons);
}